// BlockWiseSemanticAttention_944892805640
// MI455X (gfx1250) — compile-verified
//
#include <hip/hip_runtime.h>

#define Bsz 4
#define Tsz 1024
#define NT (Bsz*Tsz)   // 4096 tokens
#define NBLK 16
#define Hb 2
#define Db 2

typedef __attribute__((ext_vector_type(2))) float v2f;
typedef __attribute__((ext_vector_type(8))) float v8f;

// D = A(16x4 f32) * B(4x16 f32) + C ; wave32, fp32 matrix pipe
__device__ __forceinline__ v8f wmma4(v2f a, v2f b, v8f c) {
  return __builtin_amdgcn_wmma_f32_16x16x4_f32(false, a, false, b, (short)0, c,
                                               false, false);
}

// ---- DPP16 intra-row (16-lane half) reductions: pure VALU, no LDS ----------
template <int CTRL>
__device__ __forceinline__ float dpp_mov(float v) {
  return __int_as_float(__builtin_amdgcn_update_dpp(
      0, __float_as_int(v), CTRL, 0xF, 0xF, true));
}
__device__ __forceinline__ float halfmax(float v) {
  v = fmaxf(v, dpp_mov<0xB1>(v));   // quad_perm(1,0,3,2) = xor1
  v = fmaxf(v, dpp_mov<0x4E>(v));   // quad_perm(2,3,0,1) = xor2
  v = fmaxf(v, dpp_mov<0x141>(v));  // row_half_mirror
  v = fmaxf(v, dpp_mov<0x140>(v));  // row_mirror
  return v;
}
__device__ __forceinline__ float halfsum(float v) {
  v += dpp_mov<0xB1>(v);
  v += dpp_mov<0x4E>(v);
  v += dpp_mov<0x141>(v);
  v += dpp_mov<0x140>(v);
  return v;
}
__device__ __forceinline__ float wavesum(float v) {
  v = halfsum(v);
  v += __shfl_xor(v, 16, 32);  // single cross-half exchange (LN only)
  return v;
}

// 16x16 output tile of X[row0:+16, :Kdim] * W[col0:+16, :Kdim]^T via chained K=4 WMMAs
__device__ __forceinline__ v8f gemm_tile(const float* X, int ldx, int row0,
                                         const float* W, int ldw, int col0,
                                         int Kdim, v8f acc) {
  int lane = threadIdx.x & 31;
  int m = lane & 15;
  int hi = (lane >> 4) & 1;
  for (int k0 = 0; k0 < Kdim; k0 += 4) {
    v2f a, b;
    a.x = X[(row0 + m) * ldx + k0 + hi * 2 + 0];
    a.y = X[(row0 + m) * ldx + k0 + hi * 2 + 1];
    b.x = W[(col0 + m) * ldw + k0 + hi * 2 + 0];
    b.y = W[(col0 + m) * ldw + k0 + hi * 2 + 1];
    acc = wmma4(a, b, acc);
  }
  return acc;
}

__global__ void k_zero(float* p, int n) {
  int i = blockIdx.x * blockDim.x + threadIdx.x;
  if (i < n) p[i] = 0.f;
}

// ---- per-block QKV projection (E=4 -> 12), scale folded into Q -------------
__global__ void k_blk_qkv(const float* M, const float* Wqkv, const float* bqkv,
                          float* qb, float* kb, float* vb) {
  int id = blockIdx.x * blockDim.x + threadIdx.x;  // 16*B*T
  int blk = id >> 12;
  int bt = id & 4095;
  int b = bt >> 10, t = bt & 1023;
  int rb = blk >> 2, cb = blk & 3;
  float x[4];
  for (int j = 0; j < 4; j++) {
    int r2 = j >> 1, c2 = j & 1;
    x[j] = M[bt * 64 + (rb * 2 + r2) * 8 + cb * 2 + c2];
  }
  const float* W = Wqkv + blk * 12 * 4;
  const float* bia = bqkv + blk * 12;
  const float scale = 0.70710678118654752f;  // 1/sqrt(Dh=2)
  for (int o = 0; o < 12; o++) {
    float y = bia[o];
    for (int j = 0; j < 4; j++) y += W[o * 4 + j] * x[j];
    int e = o & 3, h = e >> 1, d = e & 1;
    int idx = (((blk * Bsz + b) * Hb + h) * Tsz + t) * Db + d;
    if (o < 4)      qb[idx] = y * scale;
    else if (o < 8) kb[idx] = y;
    else            vb[idx] = y;
  }
}

// ---- per-block flash attention, 1 wave = 1 (blk,b,qtile), both heads -------
// All global loads are unconditional (addresses clamped in-bounds); zero
// masking is a v_cndmask select, so no exec-mask divergence in the hot loop.
__global__ void k_blk_attn(const float* qb, const float* kb, const float* vb,
                           float* attn_blk) {
  __shared__ float plds[8][256];
  int lane = threadIdx.x & 31;
  int wave = threadIdx.x >> 5;
  int unit = blockIdx.x * 8 + wave;  // 16*B*64 = 4096 units
  int blk = unit >> 8;
  int b = (unit >> 6) & 3;
  int qt = unit & 63;
  int m = lane & 15;
  int hi = (lane >> 4) & 1;
  int mv = m & 1;                     // clamped V column (Dh=2)
  bool vok = (m < 2);
  float* pl = plds[wave];
  for (int h = 0; h < 2; h++) {
    const float* Q = qb + (((blk * Bsz + b) * Hb + h) * Tsz) * Db;
    const float* K = kb + (((blk * Bsz + b) * Hb + h) * Tsz) * Db;
    const float* V = vb + (((blk * Bsz + b) * Hb + h) * Tsz) * Db;
    float q0 = Q[(qt * 16 + m) * 2 + 0];
    float q1 = Q[(qt * 16 + m) * 2 + 1];
    v2f aq;
    aq.x = hi ? 0.f : q0;
    aq.y = hi ? 0.f : q1;
    float m8[8], l8[8];
    v8f accO = {};
    for (int r = 0; r < 8; r++) { m8[r] = -1.0e30f; l8[r] = 0.f; }
    for (int kt = 0; kt < 64; kt++) {
      float k0v = K[(kt * 16 + m) * 2 + 0];
      float k1v = K[(kt * 16 + m) * 2 + 1];
      v2f bk;
      bk.x = hi ? 0.f : k0v;
      bk.y = hi ? 0.f : k1v;
      v8f zero = {};
      v8f S = wmma4(aq, bk, zero);
      for (int r = 0; r < 8; r++) {
        float rm = halfmax(S[r]);
        float mn = fmaxf(m8[r], rm);
        float corr = __expf(m8[r] - mn);
        float p = __expf(S[r] - mn);
        l8[r] = l8[r] * corr + halfsum(p);
        m8[r] = mn;
        accO[r] *= corr;
        pl[(r + 8 * hi) * 16 + m] = p;  // D layout -> LDS (row, col)
      }
      asm volatile("s_wait_dscnt 0" ::: "memory");
      for (int c = 0; c < 4; c++) {     // O += P * V, K=16 in 4 chunks
        v2f ap, bv;
        ap.x = pl[m * 16 + c * 4 + hi * 2 + 0];
        ap.y = pl[m * 16 + c * 4 + hi * 2 + 1];
        int k0 = kt * 16 + c * 4 + hi * 2;
        float v0 = V[(k0 + 0) * 2 + mv];
        float v1 = V[(k0 + 1) * 2 + mv];
        bv.x = vok ? v0 : 0.f;
        bv.y = vok ? v1 : 0.f;
        accO = wmma4(ap, bv, accO);
      }
    }
    for (int r = 0; r < 8; r++) accO[r] /= l8[r];
    if (vok) {
      int t0 = qt * 16 + 8 * hi;
      for (int r = 0; r < 8; r++)
        attn_blk[((blk * Bsz + b) * Tsz + t0 + r) * 4 + h * 2 + m] = accO[r];
    }
  }
}

// ---- per-block 4x4 output projection -> all_blocks[B,T,64] -----------------
__global__ void k_blk_proj(const float* attn_blk, const float* Wo,
                           const float* bo, float* all_blocks) {
  int id = blockIdx.x * blockDim.x + threadIdx.x;  // 16*4096
  int blk = id >> 12;
  int bt = id & 4095;
  float x[4];
  for (int j = 0; j < 4; j++) x[j] = attn_blk[(blk * NT + bt) * 4 + j];
  for (int e = 0; e < 4; e++) {
    float y = bo[blk * 4 + e];
    for (int j = 0; j < 4; j++) y += Wo[(blk * 4 + e) * 4 + j] * x[j];
    all_blocks[bt * 64 + blk * 4 + e] = y;
  }
}

// ---- cross QKV GEMM [4096x64]x[64x192]^T + bias, Q scaled ------------------
__global__ void k_cross_qkv(const float* all_blocks, const float* Wqkv,
                            const float* bqkv, float* qkv) {
  int lane = threadIdx.x & 31;
  int unit = blockIdx.x * 8 + (threadIdx.x >> 5);  // 256*12 = 3072
  int mt = unit / 12, nt = unit % 12;
  int row0 = mt * 16, col0 = nt * 16;
  v8f acc = {};
  acc = gemm_tile(all_blocks, 64, row0, Wqkv, 64, col0, 64, acc);
  int n = lane & 15, hi = (lane >> 4) & 1;
  int col = col0 + n;
  float bia = bqkv[col];
  float sc = (col < 64) ? 0.35355339059327376f : 1.0f;  // 1/sqrt(Dh=8) on Q
  for (int r = 0; r < 8; r++)
    qkv[(row0 + r + 8 * hi) * 192 + col] = (acc[r] + bia) * sc;
}

// ---- cross flash attention, saves per-row (m,l) for the recv pass ----------
__global__ void k_cross_attn(const float* qkv, float* attn_c, float* ml) {
  __shared__ float plds[8][256];
  int lane = threadIdx.x & 31;
  int wave = threadIdx.x >> 5;
  int unit = blockIdx.x * 8 + wave;  // B*8*64 = 2048
  int b = unit >> 9;
  int h = (unit >> 6) & 7;
  int qt = unit & 63;
  int m = lane & 15, hi = (lane >> 4) & 1;
  int mv = m & 7;                     // clamped V column (Dh=8)
  bool vok = (m < 8);
  float* pl = plds[wave];
  const float* base = qkv + (b * Tsz) * 192;
  v2f aq[2];
  for (int c = 0; c < 2; c++) {
    const float* Qr = base + (qt * 16 + m) * 192 + h * 8;
    aq[c].x = Qr[c * 4 + hi * 2 + 0];
    aq[c].y = Qr[c * 4 + hi * 2 + 1];
  }
  float m8[8], l8[8];
  v8f accO = {};
  for (int r = 0; r < 8; r++) { m8[r] = -1.0e30f; l8[r] = 0.f; }
  for (int kt = 0; kt < 64; kt++) {
    v8f S = {};
    for (int c = 0; c < 2; c++) {  // K = Dh = 8, two chained WMMAs
      v2f bk;
      const float* Kr = base + (kt * 16 + m) * 192 + 64 + h * 8;
      bk.x = Kr[c * 4 + hi * 2 + 0];
      bk.y = Kr[c * 4 + hi * 2 + 1];
      S = wmma4(aq[c], bk, S);
    }
    for (int r = 0; r < 8; r++) {
      float rm = halfmax(S[r]);
      float mn = fmaxf(m8[r], rm);
      float corr = __expf(m8[r] - mn);
      float p = __expf(S[r] - mn);
      l8[r] = l8[r] * corr + halfsum(p);
      m8[r] = mn;
      accO[r] *= corr;
      pl[(r + 8 * hi) * 16 + m] = p;
    }
    asm volatile("s_wait_dscnt 0" ::: "memory");
    for (int c = 0; c < 4; c++) {
      v2f ap, bv;
      ap.x = pl[m * 16 + c * 4 + hi * 2 + 0];
      ap.y = pl[m * 16 + c * 4 + hi * 2 + 1];
      int k0 = kt * 16 + c * 4 + hi * 2;
      float v0 = base[(k0 + 0) * 192 + 128 + h * 8 + mv];
      float v1 = base[(k0 + 1) * 192 + 128 + h * 8 + mv];
      bv.x = vok ? v0 : 0.f;
      bv.y = vok ? v1 : 0.f;
      accO = wmma4(ap, bv, accO);
    }
  }
  for (int r = 0; r < 8; r++) accO[r] /= l8[r];
  if (m == 0) {  // lanes 0 and 16 cover rows 0-7 / 8-15
    for (int r = 0; r < 8; r++) {
      int t = qt * 16 + r + 8 * hi;
      ml[((b * 8 + h) * Tsz + t) * 2 + 0] = m8[r];
      ml[((b * 8 + h) * Tsz + t) * 2 + 1] = l8[r];
    }
  }
  if (vok) {
    for (int r = 0; r < 8; r++) {
      int t = qt * 16 + r + 8 * hi;
      attn_c[(b * Tsz + t) * 64 + h * 8 + m] = accO[r];
    }
  }
}

// ---- second pass: attention received per key = mean_h,q softmax(S)[q,k] ----
__global__ void k_cross_recv(const float* qkv, const float* ml,
                             float* attn_recv) {
  int lane = threadIdx.x & 31;
  int unit = blockIdx.x * 8 + (threadIdx.x >> 5);  // 2048
  int b = unit >> 9, h = (unit >> 6) & 7, qt = unit & 63;
  int m = lane & 15, hi = (lane >> 4) & 1;
  const float* base = qkv + (b * Tsz) * 192;
  v2f aq[2];
  for (int c = 0; c < 2; c++) {
    const float* Qr = base + (qt * 16 + m) * 192 + h * 8;
    aq[c].x = Qr[c * 4 + hi * 2 + 0];
    aq[c].y = Qr[c * 4 + hi * 2 + 1];
  }
  float m8[8], il8[8];
  for (int r = 0; r < 8; r++) {
    int t = qt * 16 + r + 8 * hi;
    m8[r] = ml[((b * 8 + h) * Tsz + t) * 2 + 0];
    il8[r] = 1.0f / ml[((b * 8 + h) * Tsz + t) * 2 + 1];
  }
  const float inv = 1.0f / (8.0f * 1024.0f);
  for (int kt = 0; kt < 64; kt++) {
    v8f S = {};
    for (int c = 0; c < 2; c++) {
      v2f bk;
      const float* Kr = base + (kt * 16 + m) * 192 + 64 + h * 8;
      bk.x = Kr[c * 4 + hi * 2 + 0];
      bk.y = Kr[c * 4 + hi * 2 + 1];
      S = wmma4(aq[c], bk, S);
    }
    float s = 0.f;
    for (int r = 0; r < 8; r++) s += __expf(S[r] - m8[r]) * il8[r];
    s += __shfl_xor(s, 16, 32);  // combine row halves -> full column sum
    if (lane < 16) atomicAdd(&attn_recv[b * Tsz + kt * 16 + lane], s * inv);
  }
}

// ---- cross output proj + residual with all_blocks --------------------------
__global__ void k_cross_proj(const float* attn_c, const float* Wo,
                             const float* bo, const float* all_blocks,
                             float* y1) {
  int lane = threadIdx.x & 31;
  int unit = blockIdx.x * 8 + (threadIdx.x >> 5);  // 256*4 = 1024
  int mt = unit >> 2, nt = unit & 3;
  int row0 = mt * 16, col0 = nt * 16;
  v8f acc = {};
  acc = gemm_tile(attn_c, 64, row0, Wo, 64, col0, 64, acc);
  int n = lane & 15, hi = (lane >> 4) & 1;
  int col = col0 + n;
  float bia = bo[col];
  for (int r = 0; r < 8; r++) {
    int row = row0 + r + 8 * hi;
    y1[row * 64 + col] = acc[r] + bia + all_blocks[row * 64 + col];
  }
}

// ---- LayerNorm(64), one wave per token, 2 elems per lane -------------------
__global__ void k_layernorm(const float* x, const float* g, const float* be,
                            float* y) {
  int lane = threadIdx.x & 31;
  int t = blockIdx.x * 8 + (threadIdx.x >> 5);  // 4096 tokens
  float v0 = x[t * 64 + lane * 2], v1 = x[t * 64 + lane * 2 + 1];
  float mu = wavesum(v0 + v1) * (1.0f / 64.0f);
  float d0 = v0 - mu, d1 = v1 - mu;
  float var = wavesum(d0 * d0 + d1 * d1) * (1.0f / 64.0f);
  float rs = rsqrtf(var + 1e-5f);
  y[t * 64 + lane * 2] = d0 * rs * g[lane * 2] + be[lane * 2];
  y[t * 64 + lane * 2 + 1] = d1 * rs * g[lane * 2 + 1] + be[lane * 2 + 1];
}

// ---- FFN layer 1: [4096x64]x[64x256]^T + exact GELU ------------------------
__global__ void k_ffn1(const float* x1, const float* W1, const float* b1,
                       float* hbuf) {
  int lane = threadIdx.x & 31;
  int unit = blockIdx.x * 8 + (threadIdx.x >> 5);  // 256*16 = 4096
  int mt = unit >> 4, nt = unit & 15;
  int row0 = mt * 16, col0 = nt * 16;
  v8f acc = {};
  acc = gemm_tile(x1, 64, row0, W1, 64, col0, 64, acc);
  int n = lane & 15, hi = (lane >> 4) & 1;
  int col = col0 + n;
  float bia = b1[col];
  for (int r = 0; r < 8; r++) {
    float z = acc[r] + bia;
    hbuf[(row0 + r + 8 * hi) * 256 + col] =
        0.5f * z * (1.0f + erff(z * 0.70710678118654752f));
  }
}

// ---- FFN layer 2: [4096x256]x[256x64]^T + residual -------------------------
__global__ void k_ffn2(const float* hbuf, const float* W2, const float* b2,
                       const float* x1, float* y2) {
  int lane = threadIdx.x & 31;
  int unit = blockIdx.x * 8 + (threadIdx.x >> 5);  // 256*4 = 1024
  int mt = unit >> 2, nt = unit & 3;
  int row0 = mt * 16, col0 = nt * 16;
  v8f acc = {};
  acc = gemm_tile(hbuf, 256, row0, W2, 256, col0, 256, acc);
  int n = lane & 15, hi = (lane >> 4) & 1;
  int col = col0 + n;
  float bia = b2[col];
  for (int r = 0; r < 8; r++) {
    int row = row0 + r + 8 * hi;
    y2[row * 64 + col] = acc[r] + bia + x1[row * 64 + col];
  }
}

// ---- sensitivity blend + from_blocks permutation ---------------------------
__global__ void k_final(const float* M, const int* tok, const float* x2,
                        const float* attn_recv, const float* semb,
                        const float* salpha, float* out) {
  int bt = blockIdx.x * blockDim.x + threadIdx.x;  // 4096
  int tk = tok[bt];
  float ar = attn_recv[bt];
  for (int i = 0; i < 16; i++) {
    float s = semb[tk * 16 + i] + ar * salpha[i];
    float sig = 1.0f / (1.0f + __expf(-s));
    int rb = i >> 2, cb = i & 3;
    for (int j = 0; j < 4; j++) {
      int r2 = j >> 1, c2 = j & 1;
      int mi = bt * 64 + (rb * 2 + r2) * 8 + cb * 2 + c2;
      float bv = M[mi];
      out[mi] = bv + (x2[bt * 64 + i * 4 + j] - bv) * sig;
    }
  }
}

extern "C" void kernel_launch(void* const* d_in, const int* in_sizes, int n_in,
                              void* d_out, int out_size, void* d_ws,
                              size_t ws_size, hipStream_t stream) {
  const float* M        = (const float*)d_in[0];
  const int*   tok      = (const int*)d_in[1];
  const float* Wqkv_blk = (const float*)d_in[2];
  const float* bqkv_blk = (const float*)d_in[3];
  const float* Wo_blk   = (const float*)d_in[4];
  const float* bo_blk   = (const float*)d_in[5];
  const float* Wqkv_c   = (const float*)d_in[6];
  const float* bqkv_c   = (const float*)d_in[7];
  const float* Wo_c     = (const float*)d_in[8];
  const float* bo_c     = (const float*)d_in[9];
  const float* W1       = (const float*)d_in[10];
  const float* b1       = (const float*)d_in[11];
  const float* W2       = (const float*)d_in[12];
  const float* b2       = (const float*)d_in[13];
  const float* g1       = (const float*)d_in[14];
  const float* be1      = (const float*)d_in[15];
  const float* g2       = (const float*)d_in[16];
  const float* be2      = (const float*)d_in[17];
  const float* semb     = (const float*)d_in[18];
  const float* salpha   = (const float*)d_in[19];

  float* ws = (float*)d_ws;
  float* attn_recv  = ws;                    // 4096
  float* qb         = attn_recv + 4096;      // 262144
  float* kb         = qb + 262144;
  float* vb         = kb + 262144;
  float* attn_blk   = vb + 262144;           // 262144
  float* all_blocks = attn_blk + 262144;     // 262144
  float* qkvc       = all_blocks + 262144;   // 786432
  float* ml         = qkvc + 786432;         // 65536
  float* attn_c     = ml + 65536;            // 262144
  float* y1         = attn_c + 262144;       // 262144
  float* x1         = y1 + 262144;           // 262144
  float* hbuf       = x1 + 262144;           // 1048576
  float* y2         = hbuf + 1048576;        // 262144
  float* x2         = y2 + 262144;           // 262144

  k_zero<<<16, 256, 0, stream>>>(attn_recv, 4096);
  k_blk_qkv<<<256, 256, 0, stream>>>(M, Wqkv_blk, bqkv_blk, qb, kb, vb);
  k_blk_attn<<<512, 256, 0, stream>>>(qb, kb, vb, attn_blk);
  k_blk_proj<<<256, 256, 0, stream>>>(attn_blk, Wo_blk, bo_blk, all_blocks);
  k_cross_qkv<<<384, 256, 0, stream>>>(all_blocks, Wqkv_c, bqkv_c, qkvc);
  k_cross_attn<<<256, 256, 0, stream>>>(qkvc, attn_c, ml);
  k_cross_recv<<<256, 256, 0, stream>>>(qkvc, ml, attn_recv);
  k_cross_proj<<<128, 256, 0, stream>>>(attn_c, Wo_c, bo_c, all_blocks, y1);
  k_layernorm<<<512, 256, 0, stream>>>(y1, g1, be1, x1);
  k_ffn1<<<512, 256, 0, stream>>>(x1, W1, b1, hbuf);
  k_ffn2<<<128, 256, 0, stream>>>(hbuf, W2, b2, x1, y2);
  k_layernorm<<<512, 256, 0, stream>>>(y2, g2, be2, x2);
  k_final<<<16, 256, 0, stream>>>(M, tok, x2, attn_recv, semb, salpha,
                                  (float*)d_out);
}